// NCEAverage_10161892622456
// MI455X (gfx1250) — compile-verified
//
#include <hip/hip_runtime.h>
#include <math.h>

// Problem constants (from reference)
#define B_    128
#define D_    128
#define OUT_  100000
#define KP1_  4097                 // K+1
#define TINV  (1.0f / 0.07f)

typedef __attribute__((ext_vector_type(2))) float v2f;
typedef __attribute__((ext_vector_type(8))) float v8f;

#define WAVES_PER_BLOCK 8
#define KBLOCKS 257                // ceil(4097/16)
#define GX 33                      // ceil(257/8)

// ---------------------------------------------------------------------------
// Kernel 1: gathered dot products via V_WMMA_F32_16X16X4_F32 (exact f32 WMMA)
// Each wave: one batch row b, 16 k-slots, both memory tables.
// A(16x4) = z1[b] chunk broadcast to all rows; B(4x16) column n = gathered row
// chunk for k-slot n.  c[0] of lane l ends up holding dot(k0 + l%16).
// ---------------------------------------------------------------------------
__global__ void nce_scores_kernel(const float* __restrict__ z1,
                                  const float* __restrict__ mem2,
                                  const float* __restrict__ mem3,
                                  const int*   __restrict__ y,
                                  const int*   __restrict__ idx,
                                  float*       __restrict__ out_s,    // [2][B][KP1]
                                  float*       __restrict__ partials) // [gy*gx][2]
{
  __shared__ __align__(16) float z1s[D_];
  __shared__ float wsum[2][WAVES_PER_BLOCK];

  const int tid = threadIdx.x;
  const int b   = blockIdx.y;
  if (tid < D_) z1s[tid] = z1[b * D_ + tid];
  __syncthreads();

  const int lane   = tid & 31;
  const int wave   = tid >> 5;
  const int kblock = blockIdx.x * WAVES_PER_BLOCK + wave;
  const int n      = lane & 15;
  const int k      = kblock * 16 + n;
  const int kc     = (k < KP1_) ? k : (KP1_ - 1);
  const bool valid = (k < KP1_) && (lane < 16);

  // idx[:,0] is replaced by the positive index y[b]
  int row = (kc == 0) ? y[b] : idx[b * KP1_ + kc];
  if (row < 0) row = 0;
  if (row >= OUT_) row = OUT_ - 1;

  const int dhalf = (lane >> 4) << 1;          // lanes 0-15 -> K=0,1 ; 16-31 -> K=2,3

  const float* mems[2] = { mem2, mem3 };
  for (int t = 0; t < 2; ++t) {
    const float* __restrict__ mrow = mems[t] + (size_t)row * D_;
    v8f c = {0.f, 0.f, 0.f, 0.f, 0.f, 0.f, 0.f, 0.f};
#pragma unroll
    for (int d0 = 0; d0 < D_; d0 += 4) {
      v2f a  = *(const v2f*)(z1s  + d0 + dhalf);   // A: z1 chunk (LDS, broadcast rows)
      v2f bb = *(const v2f*)(mrow + d0 + dhalf);   // B: gathered row chunk (column n)
      c = __builtin_amdgcn_wmma_f32_16x16x4_f32(false, a, false, bb,
                                                (short)0, c, false, false);
    }
    const float dot = c[0];
    const float s   = __expf(dot * TINV);
    if (valid) out_s[((size_t)t * B_ + b) * KP1_ + k] = s;

    // deterministic wave-level sum (lanes >=16 and invalid lanes contribute 0)
    float sv = valid ? s : 0.f;
#pragma unroll
    for (int m = 16; m > 0; m >>= 1) sv += __shfl_xor(sv, m, 32);
    if (lane == 0) wsum[t][wave] = sv;
  }
  __syncthreads();
  if (tid == 0) {
    float p0 = 0.f, p1 = 0.f;
#pragma unroll
    for (int w = 0; w < WAVES_PER_BLOCK; ++w) { p0 += wsum[0][w]; p1 += wsum[1][w]; }
    const size_t blk = (size_t)blockIdx.y * gridDim.x + blockIdx.x;
    partials[blk * 2 + 0] = p0;
    partials[blk * 2 + 1] = p1;
  }
}

// ---------------------------------------------------------------------------
// Kernel 2: deterministic reduction of block partials -> Z per table
// ---------------------------------------------------------------------------
__global__ void nce_reduceZ_kernel(const float* __restrict__ partials, int P,
                                   float* __restrict__ zz)
{
  __shared__ float red0[256];
  __shared__ float red1[256];
  const int tid = threadIdx.x;
  float a0 = 0.f, a1 = 0.f;
  for (int i = tid; i < P; i += 256) {
    a0 += partials[i * 2 + 0];
    a1 += partials[i * 2 + 1];
  }
  red0[tid] = a0; red1[tid] = a1;
  __syncthreads();
  for (int s2 = 128; s2 > 0; s2 >>= 1) {
    if (tid < s2) { red0[tid] += red0[tid + s2]; red1[tid] += red1[tid + s2]; }
    __syncthreads();
  }
  if (tid == 0) {
    const float scale = (float)OUT_ / ((float)B_ * (float)KP1_); // mean * outputSize
    zz[0] = red0[0] * scale;
    zz[1] = red1[0] * scale;
  }
}

// ---------------------------------------------------------------------------
// Kernel 3: divide scores by Z (per table)
// ---------------------------------------------------------------------------
__global__ void nce_scale_kernel(float* __restrict__ out_s,
                                 const float* __restrict__ zz)
{
  const size_t N1 = (size_t)B_ * KP1_;
  const size_t i  = (size_t)blockIdx.x * blockDim.x + threadIdx.x;
  if (i < 2 * N1) {
    const float Z = zz[(i < N1) ? 0 : 1];
    out_s[i] = out_s[i] / Z;
  }
}

// ---------------------------------------------------------------------------
// Kernel 4: stream both memory tables into the output (float4 = b128 loads)
// ---------------------------------------------------------------------------
__global__ void nce_copy_kernel(const float4* __restrict__ m2,
                                const float4* __restrict__ m3,
                                float4* __restrict__ dst)
{
  const size_t n4 = (size_t)OUT_ * D_ / 4;
  size_t i = (size_t)blockIdx.x * blockDim.x + threadIdx.x;
  const size_t stride = (size_t)gridDim.x * blockDim.x;
  for (; i < 2 * n4; i += stride) {
    dst[i] = (i < n4) ? m2[i] : m3[i - n4];
  }
}

// ---------------------------------------------------------------------------
// Kernel 5: row updates  pos = normalize(2*z - mem[y]) ; serial over b so that
// duplicate y indices resolve deterministically (last write wins), always
// reading the ORIGINAL memory (matches memory.at[y].set semantics).
// ---------------------------------------------------------------------------
__global__ void nce_update_kernel(const float* __restrict__ mem2,
                                  const float* __restrict__ mem3,
                                  const float* __restrict__ z2,
                                  const float* __restrict__ z3,
                                  const int*   __restrict__ y,
                                  float* __restrict__ dst2,
                                  float* __restrict__ dst3)
{
  const float* mem = (blockIdx.x == 0) ? mem2 : mem3;
  const float* z   = (blockIdx.x == 0) ? z2   : z3;
  float*       dst = (blockIdx.x == 0) ? dst2 : dst3;

  __shared__ float red[D_];
  __shared__ int yrow;
  const int tid = threadIdx.x;   // 0..127

  for (int b = 0; b < B_; ++b) {
    if (tid == 0) yrow = y[b];
    __syncthreads();
    const int r = yrow;
    const float m = mem[(size_t)r * D_ + tid];
    const float p = 2.0f * z[b * D_ + tid] - m;   // momentum = -1 faithfully
    red[tid] = p * p;
    __syncthreads();
    for (int s2 = 64; s2 > 0; s2 >>= 1) {
      if (tid < s2) red[tid] += red[tid + s2];
      __syncthreads();
    }
    const float nrm = sqrtf(red[0]);
    dst[(size_t)r * D_ + tid] = p / nrm;
    __syncthreads();   // protect red[] before next iteration
  }
}

// ---------------------------------------------------------------------------
extern "C" void kernel_launch(void* const* d_in, const int* in_sizes, int n_in,
                              void* d_out, int out_size, void* d_ws, size_t ws_size,
                              hipStream_t stream) {
  const float* z1   = (const float*)d_in[0];
  const float* z2   = (const float*)d_in[1];
  const float* z3   = (const float*)d_in[2];
  const float* mem2 = (const float*)d_in[3];
  const float* mem3 = (const float*)d_in[4];
  const int*   y    = (const int*)  d_in[5];
  const int*   idx  = (const int*)  d_in[6];

  float* out = (float*)d_out;
  float* ws  = (float*)d_ws;
  float* zz       = ws;        // [2] Z values
  float* partials = ws + 2;    // [GX*B_][2] block partial sums

  const size_t N1 = (size_t)B_ * KP1_;
  float* out_s  = out;                                   // I_z12 | I_z13
  float* out_m2 = out + 2 * N1;                          // new_mem2
  float* out_m3 = out_m2 + (size_t)OUT_ * D_;            // new_mem3

  // 1) gathered dots (WMMA f32) + exp, block partial sums
  dim3 g1(GX, B_);
  nce_scores_kernel<<<g1, 32 * WAVES_PER_BLOCK, 0, stream>>>(
      z1, mem2, mem3, y, idx, out_s, partials);

  // 2) deterministic Z reduction
  nce_reduceZ_kernel<<<1, 256, 0, stream>>>(partials, GX * B_, zz);

  // 3) normalize scores
  const size_t Ns = 2 * N1;
  nce_scale_kernel<<<(unsigned)((Ns + 255) / 256), 256, 0, stream>>>(out_s, zz);

  // 4) stream memory tables to output
  const size_t n4tot = (size_t)2 * OUT_ * D_ / 4;
  nce_copy_kernel<<<(unsigned)((n4tot + 255) / 256), 256, 0, stream>>>(
      (const float4*)mem2, (const float4*)mem3, (float4*)out_m2);

  // 5) overwrite updated rows (one block per table, serial over b)
  nce_update_kernel<<<2, D_, 0, stream>>>(mem2, mem3, z2, z3, y, out_m2, out_m3);
}